// EdgePredictionHead_7129645711555
// MI455X (gfx1250) — compile-verified
//
#include <hip/hip_runtime.h>
#include <hip/hip_bf16.h>

typedef __attribute__((ext_vector_type(2))) float v2f;
typedef __attribute__((ext_vector_type(4))) float v4f;
typedef __attribute__((ext_vector_type(8))) float v8f;

#define BS 65536
#define NFEAT 512
#define DIM 64

// ---------------------------------------------------------------------------
// Phase 1: GEMV via V_WMMA_F32_16X16X4_F32.
// tiles 0..4095   -> node_emb rows   (dot with W[0:64])        -> np[]
// tiles 4096..4127-> feature_emb rows(dot with W[64:128]) + b  -> fp[]
// Each wave handles one 16-row tile: 16 chained WMMAs over K=64.
// ---------------------------------------------------------------------------
__global__ void __launch_bounds__(256) proj_wmma_kernel(
    const float* __restrict__ node_emb,
    const float* __restrict__ feature_emb,
    const float* __restrict__ W,
    const float* __restrict__ bptr,
    float* __restrict__ np,
    float* __restrict__ fp)
{
    const int lane = threadIdx.x & 31;
    const int wave = threadIdx.x >> 5;
    const int tile = blockIdx.x * 8 + wave;       // 0..4127, exact

    const float* src;
    const float* w;
    float*       dst;
    int          row0;
    float        bias;
    if (tile < (BS / 16)) {
        src = node_emb;  w = W;        dst = np; row0 = tile * 16;              bias = 0.0f;
    } else {
        src = feature_emb; w = W + DIM; dst = fp; row0 = (tile - BS / 16) * 16; bias = bptr[0];
    }

    const int half = lane >> 4;   // 0: K={0,1}, 1: K={2,3}
    const int lrow = lane & 15;   // A: row M; B/D: column N

    const float* arow = src + (size_t)(row0 + lrow) * DIM + half * 2;

    v8f acc = {0.f, 0.f, 0.f, 0.f, 0.f, 0.f, 0.f, 0.f};
#pragma unroll
    for (int kc = 0; kc < DIM / 4; ++kc) {
        v2f a, bb;
        a.x  = arow[kc * 4 + 0];
        a.y  = arow[kc * 4 + 1];
        bb.x = w[kc * 4 + half * 2 + 0];   // B[k][n] = w[k] for every column n
        bb.y = w[kc * 4 + half * 2 + 1];
        acc = __builtin_amdgcn_wmma_f32_16x16x4_f32(
            /*neg_a=*/false, a, /*neg_b=*/false, bb,
            /*c_mod=*/(short)0, acc, /*reuse_a=*/false, /*reuse_b=*/false);
    }

    // Column N=0 carries the result; lanes 0 (M=0..7) and 16 (M=8..15) write.
    if (lrow == 0) {
        const int base = row0 + half * 8;
#pragma unroll
        for (int r = 0; r < 8; ++r) dst[base + r] = acc[r] + bias;
    }
}

// ---------------------------------------------------------------------------
// Phase 2: out[i][j] = np[i] + fp[j]   (fp already contains +b)
// Store-bandwidth bound: coalesced b128 non-temporal stores, fp in LDS.
// Each 256-thread block writes 16 rows (2 rows per sweep, 128 float4s/row).
// ---------------------------------------------------------------------------
__global__ void __launch_bounds__(256) bcast_add_kernel(
    const float* __restrict__ np,
    const float* __restrict__ fp,
    float* __restrict__ out)
{
    __shared__ v4f sfp[NFEAT / 4];   // 2 KB

    const int t = threadIdx.x;
    if (t < NFEAT / 4) sfp[t] = reinterpret_cast<const v4f*>(fp)[t];
    __syncthreads();

    const int ROWS = 16;
    const int row0 = blockIdx.x * ROWS;
    const int col4 = t & 127;   // float4 index within row
    const int rsub = t >> 7;    // 0 or 1

    const v4f f = sfp[col4];
    v4f* __restrict__ out4 = reinterpret_cast<v4f*>(out);

#pragma unroll
    for (int r = rsub; r < ROWS; r += 2) {
        const int   row = row0 + r;
        const float n   = np[row];
        v4f v;
        v.x = n + f.x; v.y = n + f.y; v.z = n + f.z; v.w = n + f.w;
        __builtin_nontemporal_store(v, out4 + (size_t)row * (NFEAT / 4) + col4);
    }
}

extern "C" void kernel_launch(void* const* d_in, const int* in_sizes, int n_in,
                              void* d_out, int out_size, void* d_ws, size_t ws_size,
                              hipStream_t stream) {
    const float* node_emb    = (const float*)d_in[0];
    const float* feature_emb = (const float*)d_in[1];
    const float* W           = (const float*)d_in[2];
    const float* b           = (const float*)d_in[3];
    float*       out         = (float*)d_out;

    float* ws = (float*)d_ws;
    float* np = ws;            // BS floats
    float* fp = ws + BS;       // NFEAT floats (includes +b)

    // 4096 node tiles + 32 feature tiles = 4128 wave-tiles, 8 waves/block.
    proj_wmma_kernel<<<(BS / 16 + NFEAT / 16) / 8, 256, 0, stream>>>(
        node_emb, feature_emb, W, b, np, fp);

    bcast_add_kernel<<<BS / 16, 256, 0, stream>>>(np, fp, out);
}